// MolecularGNN_38732015075832
// MI455X (gfx1250) — compile-verified
//
#include <hip/hip_runtime.h>
#include <hip/hip_bf16.h>

typedef __attribute__((ext_vector_type(2))) float v2f;
typedef __attribute__((ext_vector_type(8))) float v8f;
typedef __attribute__((ext_vector_type(4))) unsigned int v4u;
typedef __attribute__((ext_vector_type(4))) int v4i;
typedef __attribute__((ext_vector_type(8))) int v8i;

// ---------------------------------------------------------------------------
// Degree / normalization
// ---------------------------------------------------------------------------
__global__ void deg_init_kernel(float* deg, int n) {
    int i = blockIdx.x * blockDim.x + threadIdx.x;
    if (i < n) deg[i] = 1.0f;  // self-loop
}

__global__ void deg_accum_kernel(const long long* __restrict__ ei, int nedges,
                                 float* __restrict__ deg) {
    int e = blockIdx.x * blockDim.x + threadIdx.x;
    if (e < nedges) {
        int d = (int)ei[(size_t)nedges + e];  // dst row
        atomicAdd(&deg[d], 1.0f);
    }
}

__global__ void deg_to_dinv_kernel(float* deg, int n) {
    int i = blockIdx.x * blockDim.x + threadIdx.x;
    if (i < n) deg[i] = rsqrtf(deg[i]);
}

__global__ void zero_f32_kernel(float* __restrict__ p, size_t n) {
    size_t i = (size_t)blockIdx.x * blockDim.x + threadIdx.x;
    size_t stride = (size_t)gridDim.x * blockDim.x;
    for (; i < n; i += stride) p[i] = 0.0f;
}

// ---------------------------------------------------------------------------
// FP32 WMMA GEMM:  H[nrows x NCOL] = X[nrows x K] @ W[K x NCOL]
//
// The 16 x K A tile is DMA'd into LDS by the Tensor Data Mover
// (TENSOR_LOAD_TO_LDS): 2D descriptor, tile K x 16, tensor_dim1 = nrows so
// hardware zero-fills any out-of-range rows of the last tile, and hardware
// padding of 4 DWORDs per K-DWORD row gives the bank-conflict-free K+4
// LDS stride.  One wave per 16x16 output tile via V_WMMA_F32_16X16X4_F32.
//
// Per-lane WMMA operand layout (wave32, 32-bit data, ISA 7.12.2):
//   A 16x4 : lane L holds A[M=L&15][K = 2*(L>>4) + v] in VGPR v   (float2)
//   B 4x16 : lane L holds B[K = 2*(L>>4) + v][N=L&15] in VGPR v   (float2)
//   C 16x16: VGPR r, lane L -> C[M = r + 8*(L>>4)][N = L&15]      (float x8)
// ---------------------------------------------------------------------------
template <int K, int NCOL>
__global__ __launch_bounds__((NCOL / 16) * 32)
void gemm_wmma_f32_kernel(const float* __restrict__ X, const float* __restrict__ W,
                          float* __restrict__ H, int nrows) {
    constexpr int KP = K + 4;  // padded LDS row stride (floats), via TDM pad
    __shared__ float tileA[16 * KP];

    const int row0 = blockIdx.x * 16;
    const int tid  = threadIdx.x;

    // --- TDM: DMA the 16 x K fp32 tile into LDS (wave 0 issues once) -------
    if (tid < 32) {
        unsigned lds_base = (unsigned)(size_t)(void*)tileA;  // low 32b = LDS offset
        unsigned long long ga =
            (unsigned long long)(const void*)(X + (size_t)row0 * K);
        constexpr unsigned PAD_INTERVAL = (K == 128) ? 6u : 5u;  // 2^(i+1) DWORDs
        constexpr unsigned PAD_AMOUNT   = 3u;                    // 4 DWORDs pad

        v4u g0;
        g0.x = 1u;                                   // count=1 (valid user D#)
        g0.y = lds_base;                             // lds_addr
        g0.z = (unsigned)(ga & 0xFFFFFFFFu);         // global_addr[31:0]
        g0.w = (unsigned)((ga >> 32) & 0x1FFFFFFu)   // global_addr[56:32]
               | (2u << 30);                         // type = 2 ("image")

        v8i g1;
        g1[0] = (int)((2u << 16)                     // data_size = 4 bytes
                      | (1u << 20)                   // pad_enable
                      | (PAD_INTERVAL << 22) | (PAD_AMOUNT << 25));
        g1[1] = (int)(((unsigned)K & 0xFFFFu) << 16);        // tensor_dim0[15:0]
        g1[2] = (int)(((unsigned)nrows & 0xFFFFu) << 16);    // tensor_dim1[15:0]
        g1[3] = (int)(((unsigned)nrows >> 16)                // tensor_dim1[31:16]
                      | ((unsigned)K << 16));                // tile_dim0 = K
        g1[4] = 16;                                          // tile_dim1 = 16 rows
        g1[5] = K;                                           // tensor_dim0_stride
        g1[6] = 0;
        g1[7] = 0;
        v4i gz4 = {0, 0, 0, 0};            // groups 2/3 unused (2D tensor)
        v8i gz8 = {0, 0, 0, 0, 0, 0, 0, 0};

        __builtin_amdgcn_tensor_load_to_lds(g0, g1, gz4, gz4, gz8, 0);
        __builtin_amdgcn_s_wait_tensorcnt(0);
    }
    __syncthreads();
    asm volatile("" ::: "memory");  // TDM wrote LDS behind the compiler's back

    const int wave    = tid >> 5;
    const int lane    = tid & 31;
    const int m       = lane & 15;        // A row / B,C col within tile
    const int khalf   = (lane >> 4) * 2;  // 0 or 2
    const int colbase = wave * 16;

    const float* Ap = tileA + m * KP + khalf;                       // LDS
    const float* Wp = W + (size_t)khalf * NCOL + colbase + m;       // global

    v8f acc = {};
#pragma unroll 4
    for (int k = 0; k < K; k += 4) {
        v2f a;
        a.x = Ap[k];
        a.y = Ap[k + 1];
        v2f b;
        b.x = Wp[k * NCOL];
        b.y = Wp[k * NCOL + NCOL];
        acc = __builtin_amdgcn_wmma_f32_16x16x4_f32(
            /*neg_a=*/false, a, /*neg_b=*/false, b,
            /*c_mod=*/(short)0, acc, /*reuse_a=*/false, /*reuse_b=*/false);
    }

    // Store 16x16 tile: base pointer + compile-time immediate offsets.
    const int rowhalf = (lane >> 4) * 8;
    float* Hp = H + (size_t)(row0 + rowhalf) * NCOL + colbase + m;
    if (row0 + 16 <= nrows) {            // uniform fast path (always for N%16==0)
#pragma unroll
        for (int r = 0; r < 8; ++r) Hp[r * NCOL] = acc[r];
    } else {
#pragma unroll
        for (int r = 0; r < 8; ++r)
            if (row0 + rowhalf + r < nrows) Hp[r * NCOL] = acc[r];
    }
}

// ---------------------------------------------------------------------------
// Edge scatter:  agg[dst,:] += h[src,:] * dinv[src]*dinv[dst]
// One wave per edge; lanes cover features -> coalesced 128B gather segments
// and contiguous atomic segments (L2-resident accumulators on MI455X).
// ---------------------------------------------------------------------------
template <int F>
__global__ void scatter_edges_kernel(const long long* __restrict__ ei, int nedges,
                                     const float* __restrict__ h,
                                     const float* __restrict__ dinv,
                                     float* __restrict__ agg) {
    int waveId = (int)(((size_t)blockIdx.x * blockDim.x + threadIdx.x) >> 5);
    int lane   = threadIdx.x & 31;
    int nwaves = (int)(((size_t)gridDim.x * blockDim.x) >> 5);
    for (int e = waveId; e < nedges; e += nwaves) {
        int s = (int)ei[e];
        int d = (int)ei[(size_t)nedges + e];
        float norm = dinv[s] * dinv[d];
        const float* hs = h + (size_t)s * F;
        float* ad = agg + (size_t)d * F;
#pragma unroll
        for (int f = lane; f < F; f += 32) {
            atomicAdd(&ad[f], hs[f] * norm);
        }
    }
}

// ---------------------------------------------------------------------------
// agg = relu(agg + h * dinv^2 + bias)   (self-loop term fused with bias/ReLU)
// ---------------------------------------------------------------------------
template <int F>
__global__ void self_bias_relu_kernel(float* __restrict__ agg,
                                      const float* __restrict__ h,
                                      const float* __restrict__ dinv,
                                      const float* __restrict__ bias, int n) {
    size_t idx = (size_t)blockIdx.x * blockDim.x + threadIdx.x;
    if (idx >= (size_t)n * F) return;
    int i = (int)(idx / F);
    int f = (int)(idx % F);
    float di = dinv[i];
    float v = agg[idx] + h[idx] * (di * di) + bias[f];
    agg[idx] = fmaxf(v, 0.0f);
}

// ---------------------------------------------------------------------------
// out[i] = dot(h[i,0:128], Wfc) + bfc   — one wave per node, shfl reduction
// ---------------------------------------------------------------------------
__global__ void fc_out_kernel(const float* __restrict__ h,
                              const float* __restrict__ Wfc,
                              const float* __restrict__ bfc,
                              float* __restrict__ out, int n) {
    int wave = (int)(((size_t)blockIdx.x * blockDim.x + threadIdx.x) >> 5);
    int lane = threadIdx.x & 31;
    if (wave >= n) return;
    const float* hp = h + (size_t)wave * 128;
    float s = hp[lane]      * Wfc[lane]
            + hp[lane + 32] * Wfc[lane + 32]
            + hp[lane + 64] * Wfc[lane + 64]
            + hp[lane + 96] * Wfc[lane + 96];
#pragma unroll
    for (int off = 16; off > 0; off >>= 1) s += __shfl_xor(s, off, 32);
    if (lane == 0) out[wave] = s + bfc[0];
}

// ---------------------------------------------------------------------------
// Orchestration
// ---------------------------------------------------------------------------
extern "C" void kernel_launch(void* const* d_in, const int* in_sizes, int n_in,
                              void* d_out, int out_size, void* d_ws, size_t ws_size,
                              hipStream_t stream) {
    const float*     x   = (const float*)d_in[0];
    const long long* ei  = (const long long*)d_in[1];  // int64 [2, E]
    const float*     W1  = (const float*)d_in[2];      // [128, 64]
    const float*     b1  = (const float*)d_in[3];
    const float*     W2  = (const float*)d_in[4];      // [64, 128]
    const float*     b2  = (const float*)d_in[5];
    const float*     Wfc = (const float*)d_in[6];      // [128, 1]
    const float*     bfc = (const float*)d_in[7];
    float*           out = (float*)d_out;

    const int N = in_sizes[0] / 128;   // 100000
    const int E = in_sizes[1] / 2;     // 3200000

    // Workspace carve-out (a1, a2 contiguous so one zero pass covers both).
    char* ws = (char*)d_ws;
    float* dinv = (float*)ws;  ws += (size_t)N * 4;
    float* h1   = (float*)ws;  ws += (size_t)N * 64 * 4;
    float* h2   = (float*)ws;  ws += (size_t)N * 128 * 4;
    float* a1   = (float*)ws;  ws += (size_t)N * 64 * 4;
    float* a2   = (float*)ws;  ws += (size_t)N * 128 * 4;

    const int TB = 256;
    const int nodeBlocks = (N + TB - 1) / TB;
    const int edgeBlocks = (E + TB - 1) / TB;
    const int edgeWaveBlocks = (E + (TB / 32) - 1) / (TB / 32);  // wave per edge
    const int gemmBlocks = (N + 15) / 16;

    // Degrees -> dinv
    deg_init_kernel<<<nodeBlocks, TB, 0, stream>>>(dinv, N);
    deg_accum_kernel<<<edgeBlocks, TB, 0, stream>>>(ei, E, dinv);
    deg_to_dinv_kernel<<<nodeBlocks, TB, 0, stream>>>(dinv, N);

    // Zero accumulators (must happen every call; a1 and a2 are contiguous).
    zero_f32_kernel<<<4096, TB, 0, stream>>>(a1, (size_t)N * 192);

    // Layer 1: h1 = x @ W1 ; scatter ; self-loop + bias + ReLU
    gemm_wmma_f32_kernel<128, 64><<<gemmBlocks, 128, 0, stream>>>(x, W1, h1, N);
    scatter_edges_kernel<64><<<edgeWaveBlocks, TB, 0, stream>>>(ei, E, h1, dinv, a1);
    self_bias_relu_kernel<64><<<(int)(((size_t)N * 64 + TB - 1) / TB), TB, 0, stream>>>(
        a1, h1, dinv, b1, N);

    // Layer 2: h2 = a1 @ W2 ; scatter ; self-loop + bias + ReLU
    gemm_wmma_f32_kernel<64, 128><<<gemmBlocks, 256, 0, stream>>>(a1, W2, h2, N);
    scatter_edges_kernel<128><<<edgeWaveBlocks, TB, 0, stream>>>(ei, E, h2, dinv, a2);
    self_bias_relu_kernel<128><<<(int)(((size_t)N * 128 + TB - 1) / TB), TB, 0, stream>>>(
        a2, h2, dinv, b2, N);

    // Final projection
    fc_out_kernel<<<(int)(((size_t)N * 32 + TB - 1) / TB), TB, 0, stream>>>(
        a2, Wfc, bfc, out, N);
}